// dEMD_29102698397985
// MI455X (gfx1250) — compile-verified
//
#include <hip/hip_runtime.h>
#include <stdint.h>

// Greedy multi-marginal EMD (d=4, n=2048), reformulated as a fully parallel
// merge-of-prefix-sums event sweep: every sequential step of the reference
// corresponds to one prefix-sum value S_i(k); pointer states before that event
// are recoverable by binary search, so all 8192 events are independent.

#define D   4
#define N   2048
#define NWG 8
#define NT  512
#define EV_PER_WG     (D * N / NWG)      // 1024 events per workgroup
#define EV_PER_THREAD (EV_PER_WG / NT)   // 2 events per thread

// count of elements < t among S[0..2047] (nondecreasing), branch-lite binary search
__device__ __forceinline__ int count_less(const float* __restrict__ S, float t) {
    int lo = 0;
#pragma unroll
    for (int s = 2048; s > 0; s >>= 1) {
        int mid = lo + s;
        if (mid <= 2048 && S[mid - 1] < t) lo = mid;
    }
    return lo;
}
// count of elements <= t
__device__ __forceinline__ int count_leq(const float* __restrict__ S, float t) {
    int lo = 0;
#pragma unroll
    for (int s = 2048; s > 0; s >>= 1) {
        int mid = lo + s;
        if (mid <= 2048 && S[mid - 1] <= t) lo = mid;
    }
    return lo;
}

// Main kernel defined FIRST so the disasm snippet shows the async-stage prologue.
__global__ __launch_bounds__(NT) void emd_kernel_29102698397985(
        const float* __restrict__ x, float* __restrict__ out) {
    // Staged raw x, overwritten in place with inclusive prefix sums per row.
    __shared__ __align__(16) float S[D][N];     // 32 KB of the 320 KB/WGP LDS
    __shared__ float red[NT / 32];

    const int tid  = threadIdx.x;
    const int lane = tid & 31;
    const int wave = tid >> 5;

    // ---------- Phase 0: async global -> LDS stage (CDNA5 ASYNC path) ----------
    {
        uint32_t lds_base = (uint32_t)(uintptr_t)(void*)&S[0][0]; // low 32b of flat LDS addr = LDS offset
        uint64_t gbase    = (uint64_t)(uintptr_t)x;
#pragma unroll
        for (int tcp = 0; tcp < (D * N / 4) / NT; ++tcp) {        // 4 float4 chunks / thread
            uint32_t e     = (uint32_t)(tid + tcp * NT);          // float4 index in [0,2048)
            uint32_t laddr = lds_base + e * 16u;
            uint64_t gaddr = gbase + (uint64_t)e * 16u;
            // GLOBAL_LOAD_ASYNC_TO_LDS_B128: per-lane 16B memory->LDS, ASYNCcnt-tracked
            asm volatile("global_load_async_to_lds_b128 %0, %1, off"
                         :: "v"(laddr), "v"(gaddr) : "memory");
        }
        asm volatile("s_wait_asynccnt 0" ::: "memory");
    }
    __syncthreads();

    // ---------- Phase 1: in-place inclusive prefix sum per row (wave w -> row w) ----------
    if (wave < D) {
        const int row  = wave;
        const int base = lane * (N / 32);          // 64 contiguous elements per lane
        float lsum = 0.0f;
#pragma unroll 8
        for (int c = 0; c < N / 32; ++c) lsum += S[row][base + c];
        // wave32 inclusive scan of per-lane sums
        float v = lsum;
#pragma unroll
        for (int d = 1; d < 32; d <<= 1) {
            float o = __shfl_up(v, d, 32);
            if (lane >= d) v += o;
        }
        float run = v - lsum;                      // exclusive prefix for this lane
#pragma unroll 8
        for (int c = 0; c < N / 32; ++c) {
            run += S[row][base + c];
            S[row][base + c] = run;
        }
    }
    __syncthreads();

    // Terminating event: lex-first row to consume its full mass.
    float T = S[0][N - 1];
    int istar = 0;
#pragma unroll
    for (int j = 1; j < D; ++j) {
        float tj = S[j][N - 1];
        if (tj < T) { T = tj; istar = j; }
    }

    // ---------- Phase 2: independent event evaluation ----------
    float acc = 0.0f;
#pragma unroll
    for (int e = 0; e < EV_PER_THREAD; ++e) {
        const int m = blockIdx.x * EV_PER_WG + e * NT + tid;  // global event id, [0, 8192)
        const int i = m >> 11;                                // row of this event
        const int k = m & (N - 1);                            // 0-indexed position in row
        const float t = S[i][k];                              // event time = prefix sum value
        // include iff (t, i, k) lex <= (T, istar, N): the step that sets `done`
        // still pays its cost; later events are frozen.
        const bool included = (t < T) || (t == T && i <= istar);
        if (included) {
            int c[D];                                         // pointer state before this event
#pragma unroll
            for (int j = 0; j < D; ++j) {
                c[j] = (j == i) ? k
                     : (j <  i) ? count_leq (&S[j][0], t)     // ties of lower rows precede (first-argmin)
                                : count_less(&S[j][0], t);
            }
            float tprev = 0.0f;                               // time of lex-previous event (0 if first)
            int cmax = c[0], cmin = c[0];
#pragma unroll
            for (int j = 0; j < D; ++j) {
                cmax = c[j] > cmax ? c[j] : cmax;
                cmin = c[j] < cmin ? c[j] : cmin;
                if (c[j] > 0) tprev = fmaxf(tprev, S[j][c[j] - 1]);
            }
            acc += (t - tprev) * (float)(cmax - cmin);        // minval * (max(idx)-min(idx))
        }
    }

    // ---------- Reduction: wave32 shuffle + LDS + one atomic per WG ----------
#pragma unroll
    for (int d = 16; d >= 1; d >>= 1) acc += __shfl_down(acc, d, 32);
    if (lane == 0) red[wave] = acc;
    __syncthreads();
    if (tid == 0) {
        float s = 0.0f;
#pragma unroll
        for (int w = 0; w < NT / 32; ++w) s += red[w];
        atomicAdd(out, s);
    }
}

__global__ void emd_zero_29102698397985(float* out) { out[0] = 0.0f; }

extern "C" void kernel_launch(void* const* d_in, const int* in_sizes, int n_in,
                              void* d_out, int out_size, void* d_ws, size_t ws_size,
                              hipStream_t stream) {
    const float* x = (const float*)d_in[0];   // f32[4, 2048]
    float* out = (float*)d_out;               // f32 scalar
    (void)in_sizes; (void)n_in; (void)out_size; (void)d_ws; (void)ws_size;
    emd_zero_29102698397985<<<1, 1, 0, stream>>>(out);
    emd_kernel_29102698397985<<<NWG, NT, 0, stream>>>(x, out);
}